// ReadUnit_37769942401448
// MI455X (gfx1250) — compile-verified
//
#include <hip/hip_runtime.h>
#include <hip/hip_bf16.h>
#include <math.h>

typedef __attribute__((ext_vector_type(2))) float v2f;
typedef __attribute__((ext_vector_type(8))) float v8f;

#define BB 256
#define DD 512
#define SS 196

// ---------------------------------------------------------------------------
// Kernel 1: u[b,e] = ctrl[b,e]*w_attn[e];  c[b] = sum_e u[b,e]*b_concat[e]
// grid = 256 blocks (one per batch), 256 threads (each handles 2 of d=512)
// ---------------------------------------------------------------------------
__global__ void k1_u_c(const float* __restrict__ ctrl,
                       const float* __restrict__ w_attn,
                       const float* __restrict__ b_concat,
                       float* __restrict__ u,
                       float* __restrict__ c) {
  __shared__ float red[256];
  const int b = blockIdx.x, t = threadIdx.x;
  const int e1 = t, e2 = t + 256;
  const float u1 = ctrl[b * DD + e1] * w_attn[e1];
  const float u2 = ctrl[b * DD + e2] * w_attn[e2];
  u[b * DD + e1] = u1;
  u[b * DD + e2] = u2;
  red[t] = fmaf(u1, b_concat[e1], u2 * b_concat[e2]);
  __syncthreads();
  for (int off = 128; off > 0; off >>= 1) {
    if (t < off) red[t] += red[t + off];
    __syncthreads();
  }
  if (t == 0) c[b] = red[0];
}

// ---------------------------------------------------------------------------
// Kernel 2: v = u @ W_concat   ([256,512] x [512,512], fp32 WMMA 16x16x4)
// fused epilogue: g[b,d] = memory[b,d]*v[b,d] + u[b,d]
// One wave per 16x16 output tile. 512 tiles = 128 blocks x 4 waves.
// A layout (16x4): lanes 0-15 hold M rows; VGPR0/1 = K {0,1} (low half) or
// {2,3} (high half). B layout (4x16): row striped across lanes per VGPR.
// ---------------------------------------------------------------------------
__global__ void k2_gemm_wmma(const float* __restrict__ u,
                             const float* __restrict__ W,
                             const float* __restrict__ memory,
                             float* __restrict__ g) {
  const int wave = (blockIdx.x * blockDim.x + threadIdx.x) >> 5;
  const int lane = threadIdx.x & 31;
  const int m0 = (wave >> 5) << 4;   // 16 tiles along M (rows of u)
  const int n0 = (wave & 31) << 4;   // 32 tiles along N (cols of W)
  const int half = lane >> 4;
  const int lm = lane & 15;

  v8f acc = {};
  const int arow = (m0 + lm) * DD;

#pragma unroll 4
  for (int k = 0; k < DD; k += 4) {
    const int ka = k + 2 * half;                 // K index pair base per half
    v2f a = *(const v2f*)(u + arow + ka);        // A[m, ka], A[m, ka+1]
    v2f bmat;
    bmat.x = W[ka * DD + n0 + lm];               // B[ka,   n]
    bmat.y = W[(ka + 1) * DD + n0 + lm];         // B[ka+1, n]
    acc = __builtin_amdgcn_wmma_f32_16x16x4_f32(
        /*neg_a=*/false, a, /*neg_b=*/false, bmat,
        /*c_mod=*/(short)0, acc, /*reuse_a=*/false, /*reuse_b=*/false);
  }

#pragma unroll
  for (int r = 0; r < 8; ++r) {
    const int row = m0 + r + 8 * half;           // C/D layout: VGPR r -> M=r (+8 hi half)
    const int col = n0 + lm;
    const int idx = row * DD + col;
    g[idx] = fmaf(memory[idx], acc[r], u[idx]);  // g = memory*v + u
  }
}

// ---------------------------------------------------------------------------
// Kernel 3: per-batch  rai -> softmax -> out[b,e] = sum_s rvi[s]*kb[b,e,s]
// grid = 256 blocks (one per batch), 256 threads.
// Pass 1: lanes index s (contiguous in memory) -> coalesced strided-d reads.
// Pass 2: wave-cooperative dot over s per output row e, shfl_xor reduce.
// kb_proj[b] (401 KB) is re-read from L2 (all 103 MB of kb fit in 192 MB L2).
// ---------------------------------------------------------------------------
__global__ void k3_attn(const float* __restrict__ kb,
                        const float* __restrict__ g,
                        const float* __restrict__ c,
                        float* __restrict__ out) {
  __shared__ float gs[DD];
  __shared__ float red[256];
  __shared__ float rvi[256];

  const int b = blockIdx.x, t = threadIdx.x;
  const float* __restrict__ kbb = kb + (size_t)b * DD * SS;

  gs[t]       = g[b * DD + t];
  gs[t + 256] = g[b * DD + t + 256];
  __syncthreads();

  // ---- logits: rai[s] = sum_d gs[d]*kb[b,d,s] + c[b] ----
  float rai = -1e30f;
  float ex  = 0.0f;
  if (t < SS) {
    float dot = 0.0f;
#pragma unroll 8
    for (int d = 0; d < DD; ++d)
      dot = fmaf(gs[d], kbb[d * SS + t], dot);
    rai = dot + c[b];
  }

  // ---- softmax over s (block reduce max, then sum) ----
  red[t] = rai;
  __syncthreads();
  for (int off = 128; off > 0; off >>= 1) {
    if (t < off) red[t] = fmaxf(red[t], red[t + off]);
    __syncthreads();
  }
  const float mx = red[0];
  __syncthreads();

  ex = (t < SS) ? __expf(rai - mx) : 0.0f;
  red[t] = ex;
  __syncthreads();
  for (int off = 128; off > 0; off >>= 1) {
    if (t < off) red[t] += red[t + off];
    __syncthreads();
  }
  const float inv = 1.0f / red[0];
  __syncthreads();

  rvi[t] = ex * inv;
  __syncthreads();

  // ---- out[b,e] = sum_s rvi[s]*kb[b,e,s] : one wave per row e (mod 8) ----
  const int wave = t >> 5, lane = t & 31;
  for (int e = wave; e < DD; e += 8) {
    float p = 0.0f;
    for (int s = lane; s < SS; s += 32)
      p = fmaf(rvi[s], kbb[e * SS + s], p);
#pragma unroll
    for (int off = 16; off > 0; off >>= 1)
      p += __shfl_xor(p, off, 32);
    if (lane == 0) out[b * DD + e] = p;
  }
}

// ---------------------------------------------------------------------------
extern "C" void kernel_launch(void* const* d_in, const int* in_sizes, int n_in,
                              void* d_out, int out_size, void* d_ws, size_t ws_size,
                              hipStream_t stream) {
  const float* memory   = (const float*)d_in[0];  // [256,512]
  const float* ctrl     = (const float*)d_in[1];  // [256,512]
  const float* kb       = (const float*)d_in[2];  // [256,512,196]
  const float* W        = (const float*)d_in[3];  // [512,512]
  const float* b_concat = (const float*)d_in[4];  // [512]
  const float* w_attn   = (const float*)d_in[5];  // [512]
  float* out = (float*)d_out;                     // [256,512]

  float* ws = (float*)d_ws;
  float* u  = ws;                 // 131072 floats
  float* g  = ws + 131072;        // 131072 floats
  float* c  = ws + 262144;        // 256 floats

  k1_u_c<<<BB, 256, 0, stream>>>(ctrl, w_attn, b_concat, u, c);
  k2_gemm_wmma<<<128, 128, 0, stream>>>(u, W, memory, g);
  k3_attn<<<BB, 256, 0, stream>>>(kb, g, c, out);
}